// AVG_wrapper_61993557950544
// MI455X (gfx1250) — compile-verified
//
#include <hip/hip_runtime.h>

typedef __attribute__((ext_vector_type(2))) float v2f;
typedef __attribute__((ext_vector_type(8))) float v8f;

#define BATCH 64
#define SEQL  4096
#define DIM   256
#define SPLIT 8
#define STRIP (SEQL / SPLIT)   // 512 rows per block

// Masked column-sum over a 512-row strip of one batch, via fp32 WMMA.
// Each wave owns 32 D-columns; each WMMA step consumes 4 L-rows x 16 cols.
// A = all-ones 16x4, so D[m,n] = running sum of everything placed in column n.
// Loads are non-temporal: the 128-256MB input is streamed exactly once, so
// polluting the 192MB L2 with dead lines is pure loss.
__global__ __launch_bounds__(256) void masked_sum_wmma(
    const float* __restrict__ in, const int* __restrict__ length,
    float* __restrict__ ws) {
  const int b    = blockIdx.x % BATCH;
  const int s    = blockIdx.x / BATCH;
  const int tid  = threadIdx.x;
  const int wave = tid >> 5;          // wave32
  const int lane = tid & 31;
  const int n    = lane & 15;         // WMMA column within this wave's 16-col pair
  const int grp  = lane >> 4;         // half-wave selects which 2 of the 4 rows
  const int dbase = wave * 32;        // 8 waves x 32 cols = 256 = DIM

  const int len = length[b];
  const int l0  = s * STRIP;
  const int l1  = min(l0 + STRIP, len);

  // lane's fixed D offset: two adjacent columns dbase+2n, dbase+2n+1
  const float* base = in + (size_t)b * SEQL * DIM + dbase + 2 * n;

  v8f acc0 = {};                 // accumulates D columns dbase + 2n
  v8f acc1 = {};                 // accumulates D columns dbase + 2n + 1
  const v2f ones = {1.0f, 1.0f}; // A matrix: 16x4 of 1.0 (2 VGPRs/lane)
  const v2f zero = {0.0f, 0.0f};

  #pragma unroll 4
  for (int l = l0; l < l1; l += 4) {
    const int r0 = l + grp * 2;                 // this half-wave: rows r0, r0+1
    const float* p = base + (size_t)r0 * DIM;
    v2f q0 = __builtin_nontemporal_load((const v2f*)(p));        // row r0
    v2f q1 = __builtin_nontemporal_load((const v2f*)(p + DIM));  // row r0+1
    // ragged-boundary mask (bulk chunks: condition uniformly true -> cheap cndmask)
    q0 = (r0     < len) ? q0 : zero;
    q1 = (r0 + 1 < len) ? q1 : zero;
    // regroup so each WMMA column carries a single D index
    v2f b0 = {q0.x, q1.x};   // both D-offset dbase+2n   (two different rows)
    v2f b1 = {q0.y, q1.y};   // both D-offset dbase+2n+1
    acc0 = __builtin_amdgcn_wmma_f32_16x16x4_f32(false, ones, false, b0,
                                                 (short)0, acc0, false, false);
    acc1 = __builtin_amdgcn_wmma_f32_16x16x4_f32(false, ones, false, b1,
                                                 (short)0, acc1, false, false);
  }

  // D layout: VGPR0, lanes 0-15 hold M=0, N=0-15; all M rows are identical
  // (A is all-ones), so row 0 is the column sum. Lanes 0-15 store 2 floats (b64).
  if (grp == 0) {
    float* o = ws + ((size_t)s * BATCH + b) * DIM + dbase + 2 * n;
    v2f r = {acc0[0], acc1[0]};
    __builtin_nontemporal_store(r, (v2f*)o);
  }
}

// Combine the SPLIT strip partials and divide by length.
__global__ __launch_bounds__(256) void finalize(
    const float* __restrict__ ws, const int* __restrict__ length,
    float* __restrict__ out) {
  const int idx = blockIdx.x * blockDim.x + threadIdx.x;
  if (idx >= BATCH * DIM) return;
  const int b = idx / DIM;
  const int d = idx % DIM;
  float sum = 0.0f;
  #pragma unroll
  for (int s = 0; s < SPLIT; ++s)
    sum += ws[((size_t)s * BATCH + b) * DIM + d];
  out[idx] = sum / (float)length[b];
}

extern "C" void kernel_launch(void* const* d_in, const int* in_sizes, int n_in,
                              void* d_out, int out_size, void* d_ws, size_t ws_size,
                              hipStream_t stream) {
  const float* in     = (const float*)d_in[0];
  const int*   length = (const int*)d_in[1];
  float*       out    = (float*)d_out;
  float*       ws     = (float*)d_ws;   // needs SPLIT*BATCH*DIM*4 = 512 KiB

  masked_sum_wmma<<<BATCH * SPLIT, 256, 0, stream>>>(in, length, ws);
  finalize<<<(BATCH * DIM + 255) / 256, 256, 0, stream>>>(ws, length, out);
}